// LAPCycleSolverModule_42872363548790
// MI455X (gfx1250) — compile-verified
//
#include <hip/hip_runtime.h>

#define NPTS 512
#define DDIM 8192
#define EPSF 1e-6f

typedef float v2f __attribute__((ext_vector_type(2)));
typedef float v8f __attribute__((ext_vector_type(8)));

// ---------------------------------------------------------------------------
// Kernel 1: inverse L2 norms of each row of each input.
// grid = (512, 3), block = 256. inv[mat*512 + row] = 1 / max(||row||, eps)
// ---------------------------------------------------------------------------
__global__ __launch_bounds__(256) void inv_norm_kernel(
    const float* __restrict__ u0, const float* __restrict__ u1,
    const float* __restrict__ u2, float* __restrict__ inv) {
  const int row = blockIdx.x;
  const int mat = blockIdx.y;
  const float* src = (mat == 0) ? u0 : ((mat == 1) ? u1 : u2);
  const float* p = src + (size_t)row * DDIM;
  float s = 0.0f;
  for (int k = threadIdx.x; k < DDIM; k += 256) {
    float x = p[k];
    s += x * x;
  }
  __shared__ float red[256];
  red[threadIdx.x] = s;
  __syncthreads();
  for (int off = 128; off > 0; off >>= 1) {
    if (threadIdx.x < off) red[threadIdx.x] += red[threadIdx.x + off];
    __syncthreads();
  }
  if (threadIdx.x == 0) {
    float nrm = sqrtf(red[0]);
    inv[mat * NPTS + row] = 1.0f / fmaxf(nrm, EPSF);
  }
}

// ---------------------------------------------------------------------------
// Kernel 2: cost matrices via fp32 WMMA (V_WMMA_F32_16X16X4_F32).
// cost[m][i][j] = 1 - dot(A_i, B_j) * invA[i] * invB[j]
// Pairs: m0:(u0,u1)  m1:(u1,u2)  m2:(u2,u0)
// Each wave: one 16-row M-tile x four 16-col N-tiles (A reused across 4 WMMAs).
// jobs = 3 mats * 32 Mtiles * 8 Ngroups = 768; 8 waves/block -> 96 blocks.
// ---------------------------------------------------------------------------
__global__ __launch_bounds__(256) void cost_wmma_kernel(
    const float* __restrict__ u0, const float* __restrict__ u1,
    const float* __restrict__ u2, const float* __restrict__ inv,
    float* __restrict__ cost) {
  const int tid = threadIdx.x;
  const int wave = tid >> 5;
  const int lane = tid & 31;
  const int job = blockIdx.x * 8 + wave;  // 0..767
  const int mat = job >> 8;               // 0..2
  const int t = job & 255;
  const int tileM = t >> 3;  // 0..31
  const int ngrp = t & 7;    // 0..7  (4 N-tiles = 64 cols each)

  const float* A;
  const float* B;
  if (mat == 0) { A = u0; B = u1; }
  else if (mat == 1) { A = u1; B = u2; }
  else { A = u2; B = u0; }
  const int AmatIdx = mat;
  const int BmatIdx = (mat == 2) ? 0 : (mat + 1);

  const int half = lane >> 4;   // 0: K{0,1}, 1: K{2,3}
  const int r16 = lane & 15;    // M (for A) / N (for B) within tile
  const int ncol0 = ngrp * 64;

  const float* pA = A + (size_t)(tileM * 16 + r16) * DDIM + 2 * half;
  const float* pB0 = B + (size_t)(ncol0 + 0 * 16 + r16) * DDIM + 2 * half;
  const float* pB1 = B + (size_t)(ncol0 + 1 * 16 + r16) * DDIM + 2 * half;
  const float* pB2 = B + (size_t)(ncol0 + 2 * 16 + r16) * DDIM + 2 * half;
  const float* pB3 = B + (size_t)(ncol0 + 3 * 16 + r16) * DDIM + 2 * half;

  v8f acc0 = {0.f, 0.f, 0.f, 0.f, 0.f, 0.f, 0.f, 0.f};
  v8f acc1 = acc0, acc2 = acc0, acc3 = acc0;

#pragma unroll 4
  for (int k0 = 0; k0 < DDIM; k0 += 4) {
    v2f a = *(const v2f*)(pA + k0);
    v2f b0 = *(const v2f*)(pB0 + k0);
    v2f b1 = *(const v2f*)(pB1 + k0);
    v2f b2 = *(const v2f*)(pB2 + k0);
    v2f b3 = *(const v2f*)(pB3 + k0);
    acc0 = __builtin_amdgcn_wmma_f32_16x16x4_f32(false, a, false, b0, (short)0,
                                                 acc0, false, false);
    acc1 = __builtin_amdgcn_wmma_f32_16x16x4_f32(false, a, false, b1, (short)0,
                                                 acc1, false, false);
    acc2 = __builtin_amdgcn_wmma_f32_16x16x4_f32(false, a, false, b2, (short)0,
                                                 acc2, false, false);
    acc3 = __builtin_amdgcn_wmma_f32_16x16x4_f32(false, a, false, b3, (short)0,
                                                 acc3, false, false);
  }

  // Epilogue: C layout (16x16 f32): VGPR r, lanes 0-15 -> M=r, N=lane;
  // lanes 16-31 -> M=r+8, N=lane-16.
  const float* invA = inv + AmatIdx * NPTS;
  const float* invB = inv + BmatIdx * NPTS;
  float* cmat = cost + (size_t)mat * NPTS * NPTS;

  float invAr[8];
#pragma unroll
  for (int r = 0; r < 8; ++r) invAr[r] = invA[tileM * 16 + r + 8 * half];

  v8f accs[4] = {acc0, acc1, acc2, acc3};
#pragma unroll
  for (int nt = 0; nt < 4; ++nt) {
    const int Ng = ncol0 + nt * 16 + r16;
    const float ib = invB[Ng];
#pragma unroll
    for (int r = 0; r < 8; ++r) {
      const int Mg = tileM * 16 + r + 8 * half;
      cmat[(size_t)Mg * NPTS + Ng] = 1.0f - accs[nt][r] * invAr[r] * ib;
    }
  }
}

// ---------------------------------------------------------------------------
// Kernel 3: Jonker-Volgenant LAP, one 512-thread workgroup per cost matrix.
// grid = 3. Column relaxation + argmin are 512-wide parallel; duals in f64
// to match the reference solver. Writes one-hot fp32 into d_out segment.
// ---------------------------------------------------------------------------
__global__ __launch_bounds__(512) void lap_kernel(
    const float* __restrict__ cost_all, float* __restrict__ out_all) {
  const int n = NPTS;
  const int b = blockIdx.x;
  const int tid = threadIdx.x;  // one thread per column/row
  const float* cost = cost_all + (size_t)b * n * n;
  float* out = out_all + (size_t)b * n * n;

  __shared__ double u[NPTS], v[NPTS], shortest[NPTS], red_val[NPTS];
  __shared__ int red_idx[NPTS], path[NPTS], col4row[NPTS], row4col[NPTS];
  __shared__ int SC[NPTS], SR[NPTS];
  __shared__ int sh_i, sh_sink;
  __shared__ double sh_minval;

  const double INF = (double)INFINITY;

  u[tid] = 0.0;
  v[tid] = 0.0;
  col4row[tid] = -1;
  row4col[tid] = -1;
  path[tid] = -1;
  __syncthreads();

  for (int cur = 0; cur < n; ++cur) {
    shortest[tid] = INF;
    SC[tid] = 0;
    SR[tid] = 0;
    if (tid == 0) {
      sh_i = cur;
      sh_minval = 0.0;
      sh_sink = -1;
    }
    __syncthreads();

    for (;;) {
      const int i = sh_i;
      const double minv = sh_minval;
      if (tid == 0) SR[i] = 1;
      // relax column tid
      if (!SC[tid]) {
        double rr = minv + (double)cost[(size_t)i * n + tid] - u[i] - v[tid];
        if (rr < shortest[tid]) {
          shortest[tid] = rr;
          path[tid] = i;
        }
      }
      red_val[tid] = SC[tid] ? INF : shortest[tid];
      red_idx[tid] = tid;
      __syncthreads();
      // argmin over unvisited columns (tie -> smallest index, like np.argmin)
      for (int off = 256; off > 0; off >>= 1) {
        if (tid < off) {
          double v2 = red_val[tid + off];
          int i2 = red_idx[tid + off];
          if (v2 < red_val[tid] ||
              (v2 == red_val[tid] && i2 < red_idx[tid])) {
            red_val[tid] = v2;
            red_idx[tid] = i2;
          }
        }
        __syncthreads();
      }
      if (tid == 0) {
        const int j = red_idx[0];
        sh_minval = red_val[0];
        SC[j] = 1;
        if (row4col[j] == -1)
          sh_sink = j;
        else
          sh_i = row4col[j];
      }
      __syncthreads();
      if (sh_sink != -1) break;
    }

    const double minval = sh_minval;
    // dual updates (before augmentation, as in the reference)
    if (SR[tid]) {
      if (tid == cur)
        u[tid] += minval;
      else
        u[tid] += minval - shortest[col4row[tid]];
    }
    if (SC[tid]) v[tid] -= minval - shortest[tid];
    __syncthreads();
    // augment along the alternating path (scalar, thread 0)
    if (tid == 0) {
      int j = sh_sink;
      for (;;) {
        const int i = path[j];
        row4col[j] = i;
        const int t = col4row[i];
        col4row[i] = j;
        j = t;
        if (i == cur) break;
      }
    }
    __syncthreads();
  }

  // one-hot output: row tid
  float* rowp = out + (size_t)tid * n;
  const int cj = col4row[tid];
  for (int j = 0; j < n; ++j) rowp[j] = (j == cj) ? 1.0f : 0.0f;
}

// ---------------------------------------------------------------------------
extern "C" void kernel_launch(void* const* d_in, const int* in_sizes, int n_in,
                              void* d_out, int out_size, void* d_ws,
                              size_t ws_size, hipStream_t stream) {
  (void)in_sizes; (void)n_in; (void)out_size; (void)ws_size;
  const float* u0 = (const float*)d_in[0];
  const float* u1 = (const float*)d_in[1];
  const float* u2 = (const float*)d_in[2];
  float* out = (float*)d_out;

  float* cost = (float*)d_ws;                                   // 3*512*512 f32
  float* inv = (float*)((char*)d_ws + (size_t)3 * NPTS * NPTS * sizeof(float));

  dim3 gNorm(NPTS, 3);
  inv_norm_kernel<<<gNorm, 256, 0, stream>>>(u0, u1, u2, inv);

  // 768 wave-jobs / 8 waves per 256-thread block = 96 blocks
  cost_wmma_kernel<<<96, 256, 0, stream>>>(u0, u1, u2, inv, cost);

  lap_kernel<<<3, NPTS, 0, stream>>>(cost, out);
}